// Spline_73478300500440
// MI455X (gfx1250) — compile-verified
//
#include <hip/hip_runtime.h>

// Cubic B-spline eval (ORDER=4, N_CTRL=64, clamped uniform knots, 2^21 points).
// HBM-bound: 16 MB traffic -> ~0.7us @ 23.3 TB/s. Strategy: per-block, expand
// the spline into per-span cubic polynomials (61 spans) once, then the 2^21-pt
// stream is 1x b128 + 1x b64 LDS load + 3 FMAs per point. CDNA5 paths used:
// global_load_async_to_lds_b32 staging, s_wait_asynccnt, global_prefetch_b8,
// b128 streaming I/O with NT hints, wave32 blocks. WMMA is inapplicable here
// (length-4 contraction, bandwidth-bound).

typedef __attribute__((address_space(1))) int GASI;  // global int
typedef __attribute__((address_space(3))) int LASI;  // LDS int
typedef __attribute__((ext_vector_type(4))) float v4f;
typedef __attribute__((ext_vector_type(2))) float v2f;

// ---- Per-span polynomial build: symbolic de Boor in u = (x - t_lo)/w ------
// Matches the reference Cox-de Boor DP (incl. 0-denominator guards) exactly,
// but carries polynomial coefficients in u instead of point values.
__device__ void build_span_table(const float* sk, const float* sc, int m,
                                 v4f* coef4, v2f* meta) {
  const int i = m + 3;                // span: t[i] <= x < t[i+1]
  const float t_lo = sk[i];
  const float w = sk[i + 1] - t_lo;
  float P[4][4] = {};                 // P[r][k]: coeff of u^k of basis (prev level)
  float Q[4][4];
  P[0][0] = 1.0f;                     // level 0: N_i^0 = 1 on this span
  for (int j = 1; j <= 3; ++j) {
    const int first = i - j;          // first nonzero basis index at level j
    for (int r = 0; r <= j; ++r) {
      const int a = first + r;
      float acc[4] = {0.0f, 0.0f, 0.0f, 0.0f};
      if (r >= 1) {                   // (x - t[a])/(t[a+j]-t[a]) * N_a^{j-1}
        const float den = sk[a + j] - sk[a];
        if (den != 0.0f) {
          const float inv = 1.0f / den;
          const float c0 = (t_lo - sk[a]) * inv;   // (x-t[a]) = c0 + c1*u
          const float c1 = w * inv;
          for (int k = 0; k < 4; ++k) acc[k] += c0 * P[r - 1][k];
          for (int k = 1; k < 4; ++k) acc[k] += c1 * P[r - 1][k - 1];
        }
      }
      if (r <= j - 1) {               // (t[a+j+1]-x)/(t[a+j+1]-t[a+1]) * N_{a+1}^{j-1}
        const float den = sk[a + j + 1] - sk[a + 1];
        if (den != 0.0f) {
          const float inv = 1.0f / den;
          const float c0 = (sk[a + j + 1] - t_lo) * inv;  // (t[b]-x) = c0 + c1*u
          const float c1 = -w * inv;
          for (int k = 0; k < 4; ++k) acc[k] += c0 * P[r][k];
          for (int k = 1; k < 4; ++k) acc[k] += c1 * P[r][k - 1];
        }
      }
      for (int k = 0; k < 4; ++k) Q[r][k] = acc[k];
    }
    for (int r = 0; r <= j; ++r)
      for (int k = 0; k < 4; ++k) P[r][k] = Q[r][k];
  }
  // Fold control points: S(u) = sum_r c[m+r] * B_{m+r}(u)
  float a0 = 0.0f, a1 = 0.0f, a2 = 0.0f, a3 = 0.0f;
  for (int r = 0; r < 4; ++r) {
    const float c = sc[m + r];
    a0 += c * P[r][0];
    a1 += c * P[r][1];
    a2 += c * P[r][2];
    a3 += c * P[r][3];
  }
  v4f cf; cf.x = a0; cf.y = a1; cf.z = a2; cf.w = a3;
  coef4[m] = cf;
  v2f mt; mt.x = t_lo; mt.y = (w != 0.0f) ? (1.0f / w) : 0.0f;
  meta[m] = mt;
}

// ---- Hot path: span index + table lookup + Horner --------------------------
__device__ __forceinline__ float eval_pt(float x, const v4f* c4, const v2f* mt2,
                                         float scale, int mmax) {
  int m = (int)(x * scale);           // uniform interior knots: direct span index
  m = m < 0 ? 0 : (m > mmax ? mmax : m);
  const v4f cf = c4[m];               // ds_load_b128
  const v2f mt = mt2[m];              // ds_load_b64
  const float u = (x - mt.x) * mt.y;  // normalized local coordinate in [0,1]
  return fmaf(fmaf(fmaf(cf.w, u, cf.z), u, cf.y), u, cf.x);
}

__launch_bounds__(256)
__global__ void Spline_73478300500440_kernel(const float* __restrict__ x_,
                                             const float* __restrict__ knots,
                                             const float* __restrict__ ctrl,
                                             float* __restrict__ out,
                                             int n_pts, int n_knots, int n_ctrl) {
  __shared__ float s_knots[80];
  __shared__ float s_ctrl[72];
  __shared__ v4f   s_coef4[64];
  __shared__ v2f   s_meta[64];
  const int t = threadIdx.x;

  // ---- Stage tiny tables into LDS via CDNA5 async global->LDS path ----
#if defined(__gfx1250__) && __has_builtin(__builtin_amdgcn_global_load_async_to_lds_b32)
  if (t < n_knots) {
    __builtin_amdgcn_global_load_async_to_lds_b32(
        (GASI*)(knots + t), (LASI*)(s_knots + t), 0, 0);
  } else if (t < n_knots + n_ctrl) {
    const int j = t - n_knots;
    __builtin_amdgcn_global_load_async_to_lds_b32(
        (GASI*)(ctrl + j), (LASI*)(s_ctrl + j), 0, 0);
  }
#if __has_builtin(__builtin_amdgcn_s_wait_asynccnt)
  __builtin_amdgcn_s_wait_asynccnt(0);
#else
  asm volatile("s_wait_asynccnt 0" ::: "memory");
#endif
#else
  if (t < n_knots) {
    s_knots[t] = knots[t];
  } else if (t < n_knots + n_ctrl) {
    const int j = t - n_knots;
    s_ctrl[j] = ctrl[j];
  }
#endif
  __syncthreads();

  const int nspans = n_ctrl - 3;      // = 61 interior intervals
  if (t < nspans) build_span_table(s_knots, s_ctrl, t, s_coef4, s_meta);
  __syncthreads();

  const float scale = (float)nspans;
  const int   mmax  = n_ctrl - 4;     // = 60

  const long base = (long)(blockIdx.x * blockDim.x + t) * 8;
  if (base + 7 < n_pts) {
#if defined(__gfx1250__)
    __builtin_prefetch(x_ + base + 16384, 0, 0);  // global_prefetch_b8
#endif
    const v4f xa = __builtin_nontemporal_load(
        reinterpret_cast<const v4f*>(x_ + base));         // global_load_b128 NT
    const v4f xb = __builtin_nontemporal_load(
        reinterpret_cast<const v4f*>(x_ + base + 4));
    v4f ra, rb;
    ra.x = eval_pt(xa.x, s_coef4, s_meta, scale, mmax);
    ra.y = eval_pt(xa.y, s_coef4, s_meta, scale, mmax);
    ra.z = eval_pt(xa.z, s_coef4, s_meta, scale, mmax);
    ra.w = eval_pt(xa.w, s_coef4, s_meta, scale, mmax);
    rb.x = eval_pt(xb.x, s_coef4, s_meta, scale, mmax);
    rb.y = eval_pt(xb.y, s_coef4, s_meta, scale, mmax);
    rb.z = eval_pt(xb.z, s_coef4, s_meta, scale, mmax);
    rb.w = eval_pt(xb.w, s_coef4, s_meta, scale, mmax);
    __builtin_nontemporal_store(ra, reinterpret_cast<v4f*>(out + base));
    __builtin_nontemporal_store(rb, reinterpret_cast<v4f*>(out + base + 4));
  } else {
    for (long idx = base; idx < n_pts; ++idx)
      out[idx] = eval_pt(x_[idx], s_coef4, s_meta, scale, mmax);
  }
}

extern "C" void kernel_launch(void* const* d_in, const int* in_sizes, int n_in,
                              void* d_out, int out_size, void* d_ws, size_t ws_size,
                              hipStream_t stream) {
  const float* x_    = (const float*)d_in[0];  // (N_PTS,)   f32
  const float* knots = (const float*)d_in[1];  // (68,)      f32
  const float* ctrl  = (const float*)d_in[2];  // (64,1)     f32
  float* out = (float*)d_out;                  // (N_PTS,1)  f32

  const int n_pts   = in_sizes[0];
  const int n_knots = in_sizes[1];
  const int n_ctrl  = in_sizes[2];

  const int threads = 256;                     // 8 wave32 waves per block
  const int per_blk = threads * 8;             // 8 points per thread (2x float4 I/O)
  const int grid    = (n_pts + per_blk - 1) / per_blk;

  Spline_73478300500440_kernel<<<grid, threads, 0, stream>>>(
      x_, knots, ctrl, out, n_pts, n_knots, n_ctrl);
}